// SpatialAttention_44573170598437
// MI455X (gfx1250) — compile-verified
//
#include <hip/hip_runtime.h>
#include <math.h>

typedef __attribute__((ext_vector_type(2))) float v2f;
typedef __attribute__((ext_vector_type(4))) float v4f;
typedef __attribute__((ext_vector_type(8))) float v8f;

#define B_   16
#define C_   64
#define H_   224
#define W_   224
#define HW_  (H_ * W_)            // 50176
#define P_   (B_ * HW_)           // 802816 pixels
#define NTOT ((size_t)P_ * C_)    // 51380224 elements

// ---------------------------------------------------------------------------
// Kernel 1: per-pixel channel mean & max over 64 channels.
// One thread = 4 consecutive pixels (128-bit loads). Fully coalesced; reads x once.
// ---------------------------------------------------------------------------
__global__ void sa_reduce_kernel(const float* __restrict__ x,
                                 float* __restrict__ favg,
                                 float* __restrict__ fmax) {
    int i = blockIdx.x * blockDim.x + threadIdx.x;    // over P_/4
    const int hw4n = HW_ / 4;                         // 12544
    int hw4 = i % hw4n;
    int b   = i / hw4n;

    const v4f* xb = (const v4f*)x + (size_t)b * C_ * hw4n + hw4;
    v4f v = xb[0];
    v4f s = v;
    v4f m = v;
#pragma unroll 4
    for (int c = 1; c < C_; ++c) {
        v = xb[(size_t)c * hw4n];
        s.x += v.x; s.y += v.y; s.z += v.z; s.w += v.w;
        m.x = fmaxf(m.x, v.x); m.y = fmaxf(m.y, v.y);
        m.z = fmaxf(m.z, v.z); m.w = fmaxf(m.w, v.w);
    }
    const float inv = 1.0f / (float)C_;
    s.x *= inv; s.y *= inv; s.z *= inv; s.w *= inv;

    ((v4f*)favg)[(size_t)b * hw4n + hw4] = s;
    ((v4f*)fmax)[(size_t)b * hw4n + hw4] = m;
}

// ---------------------------------------------------------------------------
// Kernel 2: 7x7 conv (2ch -> 1ch, pad 3) + sigmoid, via V_WMMA_F32_16X16X4_F32.
// One wave computes 16 consecutive output pixels (one row segment).
//   M = pixel (16), K = im2col tap (98, padded to 100 = 25 chunks of 4),
//   N = output channel (only n==0 carries weights).
// A layout (16x4 f32): lane L holds M = L&15; VGPR v holds K = 2*(L>>4) + v.
// B layout (4x16 f32): lane L holds N = L&15; VGPR v holds K = 2*(L>>4) + v.
// D (16x16 f32): lane 0 holds (M=0..7, N=0) in acc[0..7]; lane 16 holds M=8..15.
// ---------------------------------------------------------------------------
__device__ __forceinline__ float sa_tap(const float* __restrict__ favg,
                                        const float* __restrict__ fmax,
                                        int b, int h, int wcol, int t) {
    // t in [0,100): tap index; >=98 is K padding.
    float r = 0.0f;
    if (t < 98) {
        int c   = t / 49;
        int rem = t - c * 49;
        int row = h + (rem / 7) - 3;
        int col = wcol + (rem % 7) - 3;
        if (row >= 0 && row < H_ && col >= 0 && col < W_) {
            const float* p = c ? fmax : favg;
            r = p[(size_t)b * HW_ + (size_t)row * W_ + col];
        }
    }
    return r;
}

__global__ void sa_conv_wmma_kernel(const float* __restrict__ favg,
                                    const float* __restrict__ fmax,
                                    const float* __restrict__ wgt,   // 98 floats
                                    const float* __restrict__ bias,  // 1 float
                                    float* __restrict__ gate) {
    const int TW   = W_ / 16;                                   // 14 tiles per row
    int lane = threadIdx.x & 31;
    int wave = (int)((blockIdx.x * blockDim.x + threadIdx.x) >> 5);  // exact grid

    int w0 = (wave % TW) * 16;
    int h  = (wave / TW) % H_;
    int b  =  wave / (TW * H_);

    int m     = lane & 15;       // M row for A / N col for B
    int khalf = lane >> 4;       // selects K pair {0,1} vs {2,3}
    int wcol  = w0 + m;          // this lane's pixel column (A rows)

    v8f acc = {0.f, 0.f, 0.f, 0.f, 0.f, 0.f, 0.f, 0.f};

#pragma unroll 5
    for (int kc = 0; kc < 25; ++kc) {
        int t0 = kc * 4 + khalf * 2;
        int t1 = t0 + 1;

        v2f a, bm;
        a[0]  = sa_tap(favg, fmax, b, h, wcol, t0);
        a[1]  = sa_tap(favg, fmax, b, h, wcol, t1);
        bm[0] = (m == 0 && t0 < 98) ? wgt[t0] : 0.0f;
        bm[1] = (m == 0 && t1 < 98) ? wgt[t1] : 0.0f;

        acc = __builtin_amdgcn_wmma_f32_16x16x4_f32(
                  /*neg_a=*/false, a, /*neg_b=*/false, bm,
                  /*c_mod=*/(short)0, acc, /*reuse_a=*/false, /*reuse_b=*/false);
    }

    float bs = bias[0];
    // Column N==0 of D lives in lane 0 (M=0..7) and lane 16 (M=8..15).
    if ((lane & 15) == 0) {
        int mbase = (lane >> 4) * 8;
        float* gp = gate + (size_t)b * HW_ + (size_t)h * W_ + w0 + mbase;
#pragma unroll
        for (int r = 0; r < 8; ++r) {
            float v = acc[r] + bs;
            gp[r] = 1.0f / (1.0f + __expf(-v));
        }
    }
}

// ---------------------------------------------------------------------------
// Kernel 3: out = sigmoid_gate * x  (gate broadcast over 64 channels).
// One thread = 4 elements; gate is L2-resident (3.2 MB). Nontemporal stores.
// ---------------------------------------------------------------------------
__global__ void sa_gate_apply_kernel(const float* __restrict__ x,
                                     const float* __restrict__ gate,
                                     float* __restrict__ out) {
    size_t i = (size_t)blockIdx.x * blockDim.x + threadIdx.x;  // over NTOT/4
    const int hw4n = HW_ / 4;
    int    hw4 = (int)(i % hw4n);
    size_t bc  = i / hw4n;
    int    b   = (int)(bc / C_);

    v4f g = ((const v4f*)gate)[(size_t)b * hw4n + hw4];
    v4f v = ((const v4f*)x)[i];
    v4f o;
    o.x = v.x * g.x; o.y = v.y * g.y; o.z = v.z * g.z; o.w = v.w * g.w;
    __builtin_nontemporal_store(o, (v4f*)out + i);
}

// ---------------------------------------------------------------------------
extern "C" void kernel_launch(void* const* d_in, const int* in_sizes, int n_in,
                              void* d_out, int out_size, void* d_ws, size_t ws_size,
                              hipStream_t stream) {
    const float* x      = (const float*)d_in[0];   // [16,64,224,224]
    const float* conv_w = (const float*)d_in[1];   // [1,2,7,7] = 98
    const float* conv_b = (const float*)d_in[2];   // [1]
    float* out = (float*)d_out;

    float* favg = (float*)d_ws;        // P_ floats
    float* fmax = favg + P_;           // P_ floats
    float* gate = fmax + P_;           // P_ floats  (total ~9.2 MB of ws)

    // Kernel 1: P_/4 = 200704 threads -> 784 blocks of 256
    sa_reduce_kernel<<<(P_ / 4) / 256, 256, 0, stream>>>(x, favg, fmax);

    // Kernel 2: one wave per 16-pixel row segment: 16*224*14 = 50176 waves
    //           = 6272 blocks of 256 (8 waves/block), exact.
    sa_conv_wmma_kernel<<<(B_ * H_ * (W_ / 16)) / 8, 256, 0, stream>>>(
        favg, fmax, conv_w, conv_b, gate);

    // Kernel 3: NTOT/4 = 12845056 threads -> 50176 blocks of 256, exact.
    sa_gate_apply_kernel<<<(int)(NTOT / 4 / 256), 256, 0, stream>>>(x, gate, out);
}